// RNNTAlignDistillLoss_75548474737117
// MI455X (gfx1250) — compile-verified
//
#include <hip/hip_runtime.h>
#include <hip/hip_bf16.h>
#include <math.h>

// ---------------------------------------------------------------------------
// RNNT align-distill loss (degenerate: only u = ylen-1 survives per batch).
//   loss = -(1/B) * sum_b [ (S_b - logZ_b * W_b) / ylen_b ]
//     S_b = sum_v soft[b,u,v] * x[v],  W_b = sum_v soft[b,u,v]
//     logZ_b = M_b + log(sum_v exp(x[v]-M_b)),  x = logits[b, t_last, u_last, :]
// One block, wave w handles batch b = w (wave32, B = 8 -> 8 waves).
// ---------------------------------------------------------------------------

typedef float v2f __attribute__((ext_vector_type(2)));
typedef float v8f __attribute__((ext_vector_type(8)));

// Full wave32 sum-reduction via two chained V_WMMA_F32_16X16X4_F32.
// Relies only on the documented A (16x4) and C/D (16x16) layouts; B is all-ones
// in every register slot, so its (undocumented) layout is irrelevant.
//   WMMA1: D1[m][n] = 2*(p_m + p_{m+16})
//   per-lane sum of the 8 D VGPRs -> halves hold 2*S_low / 2*S_high
//   WMMA2: every element = 4*(S_low + S_high) = 4*total  (broadcast to all lanes)
__device__ __forceinline__ float wave_sum_wmma(float p) {
    v2f ones; ones[0] = 1.0f; ones[1] = 1.0f;
    v8f c = {};
    v2f a; a[0] = p; a[1] = p;
    v8f d = __builtin_amdgcn_wmma_f32_16x16x4_f32(
        /*neg_a=*/false, a, /*neg_b=*/false, ones,
        /*c_mod=*/(short)0, c, /*reuse_a=*/false, /*reuse_b=*/false);
    float q = (d[0] + d[1]) + (d[2] + d[3]) + (d[4] + d[5]) + (d[6] + d[7]);
    v2f a2; a2[0] = q; a2[1] = q;
    v8f d2 = __builtin_amdgcn_wmma_f32_16x16x4_f32(
        false, a2, false, ones, (short)0, c, false, false);
    return d2[0] * 0.25f;
}

__device__ __forceinline__ float wave_max(float m) {
#pragma unroll
    for (int off = 16; off > 0; off >>= 1)
        m = fmaxf(m, __shfl_xor(m, off, 32));
    return m;
}

__global__ void __launch_bounds__(256)
rnnt_align_distill_loss_kernel(const float* __restrict__ logits,
                               const float* __restrict__ soft_labels,
                               const int* __restrict__ aligns,
                               const int* __restrict__ ylens,
                               float* __restrict__ out,
                               int B, int T, int U, int V) {
    const int wave   = threadIdx.x >> 5;
    const int lane   = threadIdx.x & 31;
    const int nwaves = blockDim.x >> 5;
    __shared__ float partial[32];

    float acc = 0.0f;  // per-wave accumulated loss_b
    for (int b = wave; b < B; b += nwaves) {   // wave-uniform loop: EXEC stays full
        const int ylen = ylens[b];
        const int u    = ylen - 1;
        const int t    = aligns[(long long)b * U + u];
        const float* __restrict__ xrow =
            logits + ((((long long)b * T + t) * U + u) * (long long)V);
        const float* __restrict__ srow =
            soft_labels + (((long long)b * U + u) * (long long)V);

        float lmax = -3.402823466e38f, s = 0.0f, w = 0.0f;
        const bool vec4 = ((V & 127) == 0);

        // Pass 1: max, soft-dot, soft-mass (coalesced float4 gathers)
        if (vec4) {
            const float4* x4 = reinterpret_cast<const float4*>(xrow);
            const float4* s4 = reinterpret_cast<const float4*>(srow);
            const int n4 = V >> 2;
            for (int i = lane; i < n4; i += 32) {
                float4 x = x4[i], sl = s4[i];
                lmax = fmaxf(fmaxf(fmaxf(lmax, x.x), fmaxf(x.y, x.z)), x.w);
                s += sl.x * x.x + sl.y * x.y + sl.z * x.z + sl.w * x.w;
                w += (sl.x + sl.y) + (sl.z + sl.w);
            }
        } else {
            for (int i = lane; i < V; i += 32) {
                float x = xrow[i], sl = srow[i];
                lmax = fmaxf(lmax, x);
                s += sl * x;
                w += sl;
            }
        }
        const float M = wave_max(lmax);

        // Pass 2: sum exp(x - M); rows are 4 KB -> re-read hits WGP$/L2
        float e = 0.0f;
        if (vec4) {
            const float4* x4 = reinterpret_cast<const float4*>(xrow);
            const int n4 = V >> 2;
            for (int i = lane; i < n4; i += 32) {
                float4 x = x4[i];
                e += expf(x.x - M) + expf(x.y - M) + expf(x.z - M) + expf(x.w - M);
            }
        } else {
            for (int i = lane; i < V; i += 32) e += expf(xrow[i] - M);
        }

        // Wave-wide sums via the WMMA reduction chain (result broadcast to all lanes)
        const float E = wave_sum_wmma(e);
        const float S = wave_sum_wmma(s);
        const float W = wave_sum_wmma(w);

        const float logZ = M + logf(E);
        acc += (S - logZ * W) / (float)ylen;
    }

    if (lane == 0) partial[wave] = acc;
    __syncthreads();
    if (threadIdx.x == 0) {
        float tot = 0.0f;
        for (int i = 0; i < nwaves; ++i) tot += partial[i];
        out[0] = -tot / (float)B;
    }
}

extern "C" void kernel_launch(void* const* d_in, const int* in_sizes, int n_in,
                              void* d_out, int out_size, void* d_ws, size_t ws_size,
                              hipStream_t stream) {
    // setup_inputs order: logits, ys, soft_labels, aligns, xlens, ylens
    const float* logits      = (const float*)d_in[0];
    const float* soft_labels = (const float*)d_in[2];
    const int*   aligns      = (const int*)d_in[3];
    const int*   ylens       = (const int*)d_in[5];
    float*       out         = (float*)d_out;

    const int B = in_sizes[4];                       // xlens: [B]
    const int U = in_sizes[1] / B;                   // ys:    [B,U]
    const int V = in_sizes[2] / (in_sizes[1]);       // soft:  [B,U,V] -> V = |soft|/(B*U)
    const int T = (int)((long long)in_sizes[0] / ((long long)B * U * V));

    // Latency-bound problem (~64 KB of gathered reads): one launch, one block,
    // 8 wave32s -> one wave per batch element, fully fused.
    rnnt_align_distill_loss_kernel<<<1, 256, 0, stream>>>(
        logits, soft_labels, aligns, ylens, out, B, T, U, V);
}